// _ASTGCNBlock_86655260164724
// MI455X (gfx1250) — compile-verified
//
#include <hip/hip_runtime.h>
#include <hip/hip_bf16.h>
#include <math.h>

// ---------------- problem constants ----------------
#define BB 8
#define NN 2000
#define CC 32
#define TT 12
#define KK 3
#define FCH 64
#define FTM 64
#define QQ (CC * TT)        // 384, the (c,t) flattened dim
#define NN2 ((size_t)NN * NN)

typedef __attribute__((ext_vector_type(2))) float v2f;
typedef __attribute__((ext_vector_type(8))) float v8f;

__device__ __forceinline__ float sigmoidf_(float v) {
  return 1.0f / (1.0f + __expf(-v));
}

#define WMMA_F32_16x16x4(a, b, c) \
  __builtin_amdgcn_wmma_f32_16x16x4_f32(false, (a), false, (b), (short)0, (c), false, false)

// ================= temporal attention (tiny) =================

// lhs1[b,t,c] = sum_n x[b,n,c,t] * U1[n]
__global__ void k_lhs1(const float* __restrict__ x, const float* __restrict__ U1,
                       float* __restrict__ lhs1) {
  int idx = blockIdx.x * blockDim.x + threadIdx.x;
  if (idx >= BB * TT * CC) return;
  int b = idx / (TT * CC);
  int r = idx % (TT * CC);
  int t = r / CC, c = r % CC;
  float s = 0.f;
  const float* xp = x + ((size_t)b * NN * CC + c) * TT + t;
  for (int n = 0; n < NN; ++n) s += xp[(size_t)n * CC * TT] * U1[n];
  lhs1[idx] = s;
}

// lhs2[b,t,n] = sum_c lhs1[b,t,c] * U2[c,n]
__global__ void k_lhs2(const float* __restrict__ lhs1, const float* __restrict__ U2,
                       float* __restrict__ lhs2) {
  int idx = blockIdx.x * blockDim.x + threadIdx.x;
  if (idx >= BB * TT * NN) return;
  int b = idx / (TT * NN);
  int r = idx % (TT * NN);
  int t = r / NN, n = r % NN;
  float s = 0.f;
  const float* lp = lhs1 + (b * TT + t) * CC;
  for (int c = 0; c < CC; ++c) s += lp[c] * U2[c * NN + n];
  lhs2[(b * TT + t) * NN + n] = s;
}

// rhsT[b,n,t] = sum_c U3[c] * x[b,n,c,t]
__global__ void k_rhsT(const float* __restrict__ x, const float* __restrict__ U3,
                       float* __restrict__ rhsT) {
  int idx = blockIdx.x * blockDim.x + threadIdx.x;
  if (idx >= BB * NN * TT) return;
  int b = idx / (NN * TT);
  int r = idx % (NN * TT);
  int n = r / TT, t = r % TT;
  float s = 0.f;
  const float* xp = x + ((size_t)(b * NN + n) * CC) * TT + t;
  for (int c = 0; c < CC; ++c) s += U3[c] * xp[c * TT];
  rhsT[(b * NN + n) * TT + t] = s;
}

// E1[b,i,j] = sigmoid( sum_n lhs2[b,i,n]*rhsT[b,n,j] + be[i,j] )
__global__ void k_E1(const float* __restrict__ lhs2, const float* __restrict__ rhsT,
                     const float* __restrict__ be, float* __restrict__ E1) {
  int idx = blockIdx.x * blockDim.x + threadIdx.x;
  if (idx >= BB * TT * TT) return;
  int b = idx / (TT * TT);
  int r = idx % (TT * TT);
  int i = r / TT, j = r % TT;
  float s = 0.f;
  const float* lp = lhs2 + (b * TT + i) * NN;
  const float* rp = rhsT + (size_t)b * NN * TT + j;
  for (int n = 0; n < NN; ++n) s += lp[n] * rp[n * TT];
  E1[idx] = sigmoidf_(s + be[i * TT + j]);
}

// tAt[b,:,j] = softmax_i( sum_k Ve[i,k]*E1[b,k,j] )
__global__ void k_tsoftmax(const float* __restrict__ E1, const float* __restrict__ Ve,
                           float* __restrict__ tAt) {
  int idx = blockIdx.x * blockDim.x + threadIdx.x;
  if (idx >= BB * TT) return;
  int b = idx / TT, j = idx % TT;
  float e[TT];
  float mx = -3.4e38f;
  for (int i = 0; i < TT; ++i) {
    float s = 0.f;
    for (int k = 0; k < TT; ++k) s += Ve[i * TT + k] * E1[(b * TT + k) * TT + j];
    e[i] = s;
    mx = fmaxf(mx, s);
  }
  float sum = 0.f;
  for (int i = 0; i < TT; ++i) { e[i] = __expf(e[i] - mx); sum += e[i]; }
  float inv = 1.f / sum;
  for (int i = 0; i < TT; ++i) tAt[(b * TT + i) * TT + j] = e[i] * inv;
}

// xTAt[b,n,c,t] = sum_t' x[b,n,c,t'] * tAt[b,t',t]
__global__ void k_xTAt(const float* __restrict__ x, const float* __restrict__ tAt,
                       float* __restrict__ xTAt) {
  size_t idx = (size_t)blockIdx.x * blockDim.x + threadIdx.x;
  if (idx >= (size_t)BB * NN * CC * TT) return;
  int t = (int)(idx % TT);
  size_t base = idx - t;
  int b = (int)(idx / ((size_t)NN * CC * TT));
  float s = 0.f;
  for (int tp = 0; tp < TT; ++tp) s += x[base + tp] * tAt[(b * TT + tp) * TT + t];
  xTAt[idx] = s;
}

// ================= spatial attention scores (small) =================

// tmpS[b,n,c] = sum_t xTAt[b,n,c,t] * W1[t]
__global__ void k_tmpS(const float* __restrict__ xTAt, const float* __restrict__ W1,
                       float* __restrict__ tmpS) {
  int idx = blockIdx.x * blockDim.x + threadIdx.x;
  if (idx >= BB * NN * CC) return;
  float s = 0.f;
  const float* xp = xTAt + (size_t)idx * TT;
  for (int t = 0; t < TT; ++t) s += xp[t] * W1[t];
  tmpS[idx] = s;
}

// lhsS[b,n,t] = sum_c tmpS[b,n,c] * W2[c,t]
__global__ void k_lhsS(const float* __restrict__ tmpS, const float* __restrict__ W2,
                       float* __restrict__ lhsS) {
  int idx = blockIdx.x * blockDim.x + threadIdx.x;
  if (idx >= BB * NN * TT) return;
  int t = idx % TT;
  int bn = idx / TT;
  float s = 0.f;
  const float* tp = tmpS + (size_t)bn * CC;
  for (int c = 0; c < CC; ++c) s += tp[c] * W2[c * TT + t];
  lhsS[idx] = s;
}

// rhsS[b,t,n] = sum_c W3[c] * xTAt[b,n,c,t]
__global__ void k_rhsS(const float* __restrict__ xTAt, const float* __restrict__ W3,
                       float* __restrict__ rhsS) {
  int idx = blockIdx.x * blockDim.x + threadIdx.x;
  if (idx >= BB * TT * NN) return;
  int b = idx / (TT * NN);
  int r = idx % (TT * NN);
  int t = r / NN, n = r % NN;
  float s = 0.f;
  const float* xp = xTAt + ((size_t)(b * NN + n) * CC) * TT + t;
  for (int c = 0; c < CC; ++c) s += W3[c] * xp[c * TT];
  rhsS[(b * TT + t) * NN + n] = s;
}

// P[b,j,i] = sigmoid( sum_t lhsS[b,j,t]*rhsS[b,t,i] + bs[j,i] )
__global__ void k_P(const float* __restrict__ lhsS, const float* __restrict__ rhsS,
                    const float* __restrict__ bs, float* __restrict__ P) {
  size_t idx = (size_t)blockIdx.x * blockDim.x + threadIdx.x;
  if (idx >= (size_t)BB * NN2) return;
  int i = (int)(idx % NN);
  size_t r = idx / NN;
  int j = (int)(r % NN);
  int b = (int)(r / NN);
  float s = bs[(size_t)j * NN + i];
  const float* lp = lhsS + (size_t)(b * NN + j) * TT;
  const float* rp = rhsS + (size_t)b * TT * NN + i;
  for (int t = 0; t < TT; ++t) s += lp[t] * rp[(size_t)t * NN];
  P[idx] = sigmoidf_(s);
}

// ================= LDS-tiled transpose: VsT[j,i] = Vs[i,j] =================
__global__ void k_transpose(const float* __restrict__ A, float* __restrict__ At) {
  __shared__ float tile[16][17];
  int bx = blockIdx.x * 16, by = blockIdx.y * 16;
  int tx = threadIdx.x, ty = threadIdx.y;
  tile[ty][tx] = A[(size_t)(by + ty) * NN + bx + tx];
  __syncthreads();
  At[(size_t)(bx + ty) * NN + by + tx] = tile[tx][ty];
}

// ================= big GEMM 1: S2[b] = Vs (2000x2000) @ P[b] (2000x2000) =================
// A fragments loaded from VsT so lanes are contiguous. Each wave: 16x80 strip
// (5 accumulators, 125 col-tiles = 25 groups * 5 -> NO guards in the k-loop).
__global__ void k_gemm_vs(const float* __restrict__ VsT, const float* __restrict__ P,
                          float* __restrict__ S2) {
  const int lane = threadIdx.x & 31;
  const int wave = threadIdx.x >> 5;
  const int b = blockIdx.y;
  const int NG = 25;                  // groups of 5 col-tiles (80 cols)
  int task = blockIdx.x * (blockDim.x >> 5) + wave;
  if (task >= (NN / 16) * NG) return; // wave-uniform, top-level only
  int tm = task / NG;
  int tg = task % NG;
  int row0 = tm * 16;
  int col0 = tg * 80;
  const float* Bm = P + (size_t)b * NN2;
  float* Cm = S2 + (size_t)b * NN2;
  v8f acc[5] = {};
  const int mm = lane & 15;
  const int kkl = (lane >> 4) << 1;   // 0 or 2
  const float* ap = VsT + (size_t)kkl * NN + row0 + mm;   // lane-contiguous
  const float* bp = Bm + (size_t)kkl * NN + col0 + mm;
#pragma unroll 2
  for (int k0 = 0; k0 < NN; k0 += 4) {
    v2f a;
    a.x = ap[0];
    a.y = ap[NN];
#pragma unroll
    for (int u = 0; u < 5; ++u) {
      v2f bv;
      bv.x = bp[u * 16];
      bv.y = bp[u * 16 + NN];
      acc[u] = WMMA_F32_16x16x4(a, bv, acc[u]);
    }
    ap += 4 * (size_t)NN;
    bp += 4 * (size_t)NN;
  }
  const int mb = (lane >> 4) * 8;
#pragma unroll
  for (int u = 0; u < 5; ++u)
#pragma unroll
    for (int r = 0; r < 8; ++r)
      Cm[(size_t)(row0 + mb + r) * NN + col0 + u * 16 + mm] = acc[u][r];
}

// softmax over axis=1 (row index) per column: SAt[b,:,i] = softmax(S2[b,:,i])
__global__ void k_softmax_col(const float* __restrict__ S2, float* __restrict__ SAt) {
  int b = blockIdx.y;
  int i = blockIdx.x * blockDim.x + threadIdx.x;
  if (i >= NN) return;
  const float* src = S2 + (size_t)b * NN2 + i;
  float* dst = SAt + (size_t)b * NN2 + i;
  float m = -3.4e38f, s = 0.f;
  for (int j = 0; j < NN; ++j) {
    float v = src[(size_t)j * NN];
    float nm = fmaxf(m, v);
    s = s * __expf(m - nm) + __expf(v - nm);
    m = nm;
  }
  float inv = 1.f / s;
  for (int j = 0; j < NN; ++j) {
    float v = src[(size_t)j * NN];
    dst[(size_t)j * NN] = __expf(v - m) * inv;
  }
}

// ================= big GEMM 2: H[b,k] = (cheb[k]*SAt[b])^T @ X[b] =================
// A^T[i,j] = cheb[k][j*N+i] * SAt[b][j*N+i]  (lane-contiguous over i)
// X[b] = x[b] viewed (N x 384).  H: (N x 384).  Wave: 16x64 strip, exact tiling.
__global__ void k_gemm_cheb(const float* __restrict__ cheb, const float* __restrict__ SAt,
                            const float* __restrict__ x, float* __restrict__ H) {
  const int lane = threadIdx.x & 31;
  const int wave = threadIdx.x >> 5;
  int bk = blockIdx.y;                 // 0..23
  int b = bk / KK, k = bk % KK;
  const int NG = QQ / 64;              // 6
  int task = blockIdx.x * (blockDim.x >> 5) + wave;
  if (task >= (NN / 16) * NG) return;  // wave-uniform, top-level only
  int tm = task / NG;
  int tg = task % NG;
  int row0 = tm * 16;
  const float* A1 = cheb + (size_t)k * NN2;
  const float* A2 = SAt + (size_t)b * NN2;
  const float* Bm = x + (size_t)b * NN * QQ;
  float* Cm = H + ((size_t)b * KK + k) * NN * QQ;
  v8f acc[4] = {};
  const int mm = lane & 15;
  const int kkl = (lane >> 4) << 1;
  const float* a1p = A1 + (size_t)kkl * NN + row0 + mm;
  const float* a2p = A2 + (size_t)kkl * NN + row0 + mm;
  const float* bp = Bm + (size_t)kkl * QQ + tg * 64 + mm;
#pragma unroll 2
  for (int k0 = 0; k0 < NN; k0 += 4) {
    v2f a;
    a.x = a1p[0] * a2p[0];
    a.y = a1p[NN] * a2p[NN];
#pragma unroll
    for (int u = 0; u < 4; ++u) {
      v2f bv;
      bv.x = bp[u * 16];
      bv.y = bp[u * 16 + QQ];
      acc[u] = WMMA_F32_16x16x4(a, bv, acc[u]);
    }
    a1p += 4 * (size_t)NN;
    a2p += 4 * (size_t)NN;
    bp += 4 * (size_t)QQ;
  }
  const int mb = (lane >> 4) * 8;
#pragma unroll
  for (int u = 0; u < 4; ++u)
#pragma unroll
    for (int r = 0; r < 8; ++r)
      Cm[(size_t)(row0 + mb + r) * QQ + tg * 64 + u * 16 + mm] = acc[u][r];
}

// ================= GEMM 3: gcn[b,i,f,t] = relu( sum_{k,c} H[b,k,i,c*T+t] * Theta[k,c,f] )
// per batch: M = N*T = 24000 rows (i*12+t), K = 96, N = 64.  Theta is a perfect (96x64) B.
__global__ void k_gemm_theta(const float* __restrict__ H, const float* __restrict__ Theta,
                             float* __restrict__ gcn) {
  const int lane = threadIdx.x & 31;
  const int wave = threadIdx.x >> 5;
  int b = blockIdx.y;
  const int NTM = NN * TT / 16;        // 1500
  int task = blockIdx.x * (blockDim.x >> 5) + wave;
  if (task >= NTM) return;             // wave-uniform, top-level only
  int row0 = task * 16;
  const int mm = lane & 15;
  const int kkl = (lane >> 4) << 1;
  v8f acc[4] = {};
  int rrow = row0 + mm;
  int ai = rrow / TT, at = rrow % TT;
#pragma unroll 2
  for (int k0 = 0; k0 < KK * CC; k0 += 4) {
    int cidx = k0 + kkl;               // even, so cidx+1 stays in same k-block
    int kc = cidx >> 5, cc2 = cidx & 31;
    const float* hp = H + (((size_t)b * KK + kc) * NN + ai) * QQ + cc2 * TT + at;
    v2f a;
    a.x = hp[0];
    a.y = hp[TT];
#pragma unroll
    for (int u = 0; u < 4; ++u) {
      const float* bp = Theta + (size_t)cidx * FCH + u * 16 + mm;
      v2f bv;
      bv.x = bp[0];
      bv.y = bp[FCH];
      acc[u] = WMMA_F32_16x16x4(a, bv, acc[u]);
    }
  }
  const int mb = (lane >> 4) * 8;
#pragma unroll
  for (int u = 0; u < 4; ++u) {
#pragma unroll
    for (int r = 0; r < 8; ++r) {
      int row = row0 + mb + r;
      int i2 = row / TT, t2 = row % TT;
      int f = u * 16 + mm;
      gcn[(((size_t)b * NN + i2) * FCH + f) * TT + t2] = fmaxf(acc[u][r], 0.f);
    }
  }
}

// ================= fused time-conv(1x3) + residual(1x1) + relu + LayerNorm =================
// one block (64 threads = f) per (b,n,t)
__global__ void k_out(const float* __restrict__ gcn, const float* __restrict__ x,
                      const float* __restrict__ tc_w, const float* __restrict__ tc_b,
                      const float* __restrict__ rc_w, const float* __restrict__ rc_b,
                      const float* __restrict__ ln_g, const float* __restrict__ ln_b,
                      float* __restrict__ out) {
  __shared__ float zs[FTM];
  int f = threadIdx.x;
  int idx = blockIdx.x;                // (b*N + n)*T + t
  int t = idx % TT;
  int bn = idx / TT;
  const float* g = gcn + (size_t)bn * FCH * TT;   // [fi*12 + tt]
  float acc = tc_b[f];
#pragma unroll
  for (int dt = 0; dt < 3; ++dt) {
    int tt = t + dt - 1;
    if (tt >= 0 && tt < TT) {
      const float* w = tc_w + (size_t)f * FCH * 3 + dt;
      for (int fi = 0; fi < FCH; ++fi) acc += g[fi * TT + tt] * w[fi * 3];
    }
  }
  float res = rc_b[f];
  const float* xp = x + (size_t)bn * CC * TT + t;
  const float* rw = rc_w + f * CC;
  for (int c = 0; c < CC; ++c) res += xp[c * TT] * rw[c];
  float z = fmaxf(res + acc, 0.f);
  zs[f] = z;
  __syncthreads();
  float mu = 0.f;
  for (int q = 0; q < FTM; ++q) mu += zs[q];
  mu *= (1.f / FTM);
  float var = 0.f;
  for (int q = 0; q < FTM; ++q) { float d = zs[q] - mu; var += d * d; }
  var *= (1.f / FTM);
  float zn = (z - mu) * rsqrtf(var + 1e-5f) * ln_g[f] + ln_b[f];
  out[((size_t)bn * FTM + f) * TT + t] = zn;
}

// ================= launcher =================
extern "C" void kernel_launch(void* const* d_in, const int* in_sizes, int n_in,
                              void* d_out, int out_size, void* d_ws, size_t ws_size,
                              hipStream_t stream) {
  const float* x    = (const float*)d_in[0];
  const float* cheb = (const float*)d_in[1];
  const float* U1   = (const float*)d_in[2];
  const float* U2   = (const float*)d_in[3];
  const float* U3   = (const float*)d_in[4];
  const float* be   = (const float*)d_in[5];
  const float* Ve   = (const float*)d_in[6];
  const float* W1   = (const float*)d_in[7];
  const float* W2   = (const float*)d_in[8];
  const float* W3   = (const float*)d_in[9];
  const float* bs   = (const float*)d_in[10];
  const float* Vs   = (const float*)d_in[11];
  const float* Theta= (const float*)d_in[12];
  const float* tc_w = (const float*)d_in[13];
  const float* tc_b = (const float*)d_in[14];
  const float* rc_w = (const float*)d_in[15];
  const float* rc_b = (const float*)d_in[16];
  const float* ln_g = (const float*)d_in[17];
  const float* ln_b = (const float*)d_in[18];
  float* out = (float*)d_out;

  float* ws = (float*)d_ws;
  size_t o = 0;
  float* xTAt = ws + o; o += (size_t)BB * NN * CC * TT;   // 6.14M
  float* lhs1 = ws + o; o += (size_t)BB * TT * CC;
  float* lhs2 = ws + o; o += (size_t)BB * TT * NN;
  float* rhsT = ws + o; o += (size_t)BB * NN * TT;
  float* E1   = ws + o; o += (size_t)BB * TT * TT;
  float* tAt  = ws + o; o += (size_t)BB * TT * TT;
  float* tmpS = ws + o; o += (size_t)BB * NN * CC;
  float* lhsS = ws + o; o += (size_t)BB * NN * TT;
  float* rhsS = ws + o; o += (size_t)BB * TT * NN;
  float* VsT  = ws + o; o += NN2;                         // 4M (transposed Vs)
  float* Pbuf = ws + o; o += (size_t)BB * NN2;            // 32M (reused as SAt)
  float* S2   = ws + o; o += (size_t)BB * NN2;            // 32M
  float* Hbuf = ws + o; o += (size_t)BB * KK * NN * QQ;   // 18.4M
  float* gcn  = ws + o; o += (size_t)BB * NN * FCH * TT;  // 12.3M
  float* SAt = Pbuf;  // P dead after S2 is computed

  const int TPB = 256;

  // temporal attention
  k_lhs1<<<dim3((BB*TT*CC + TPB-1)/TPB), TPB, 0, stream>>>(x, U1, lhs1);
  k_lhs2<<<dim3((BB*TT*NN + TPB-1)/TPB), TPB, 0, stream>>>(lhs1, U2, lhs2);
  k_rhsT<<<dim3((BB*NN*TT + TPB-1)/TPB), TPB, 0, stream>>>(x, U3, rhsT);
  k_E1<<<dim3((BB*TT*TT + TPB-1)/TPB), TPB, 0, stream>>>(lhs2, rhsT, be, E1);
  k_tsoftmax<<<dim3(1), 128, 0, stream>>>(E1, Ve, tAt);
  {
    size_t n = (size_t)BB * NN * CC * TT;
    k_xTAt<<<dim3((unsigned)((n + TPB-1)/TPB)), TPB, 0, stream>>>(x, tAt, xTAt);
  }

  // spatial attention
  k_tmpS<<<dim3((BB*NN*CC + TPB-1)/TPB), TPB, 0, stream>>>(xTAt, W1, tmpS);
  k_lhsS<<<dim3((BB*NN*TT + TPB-1)/TPB), TPB, 0, stream>>>(tmpS, W2, lhsS);
  k_rhsS<<<dim3((BB*TT*NN + TPB-1)/TPB), TPB, 0, stream>>>(xTAt, W3, rhsS);
  {
    size_t n = (size_t)BB * NN2;
    k_P<<<dim3((unsigned)((n + TPB-1)/TPB)), TPB, 0, stream>>>(lhsS, rhsS, bs, Pbuf);
  }

  // transpose Vs once so GEMM-1 A-fragment loads are lane-contiguous
  k_transpose<<<dim3(NN/16, NN/16), dim3(16,16), 0, stream>>>(Vs, VsT);

  // S2 = Vs @ P   (WMMA, guard-free 16x80 strips)
  {
    int tasks = (NN/16) * 25;                  // 3125 waves per batch
    k_gemm_vs<<<dim3((tasks + 7)/8, BB), TPB, 0, stream>>>(VsT, Pbuf, S2);
  }
  // column softmax -> SAt (reuses P buffer)
  k_softmax_col<<<dim3((NN + TPB-1)/TPB, BB), TPB, 0, stream>>>(S2, SAt);

  // H = (cheb*SAt)^T @ X   (WMMA)
  {
    int tasks = (NN/16) * (QQ/64);             // 750 waves per (b,k)
    k_gemm_cheb<<<dim3((tasks + 7)/8, BB*KK), TPB, 0, stream>>>(cheb, SAt, x, Hbuf);
  }
  // gcn = relu(H x Theta)  (WMMA)
  {
    int tasks = NN * TT / 16;                  // 1500 waves per batch
    k_gemm_theta<<<dim3((tasks + 7)/8, BB), TPB, 0, stream>>>(Hbuf, Theta, gcn);
  }

  // fused conv + residual + relu + LayerNorm
  k_out<<<dim3(BB*NN*TT), FTM, 0, stream>>>(gcn, x, tc_w, tc_b, rc_w, rc_b, ln_g, ln_b, out);
}